// BatchedPeriodicDistance_81149112090679
// MI455X (gfx1250) — compile-verified
//
#include <hip/hip_runtime.h>
#include <math.h>

#ifndef __has_builtin
#define __has_builtin(x) 0
#endif

// 12-byte element for the [E,3] float streams -> global_load/store_b96
struct Float3 { float x, y, z; };

#define BPD_BLOCK 256

typedef __attribute__((address_space(1))) int gl_int;
typedef __attribute__((address_space(3))) int lds_int;

__global__ __launch_bounds__(BPD_BLOCK) void bpd_edges_kernel(
    const float* __restrict__ pos,     // [N_ATOMS,3]
    const float* __restrict__ box,     // [B,3,3] flat
    const int*   __restrict__ batch,   // [N_ATOMS]
    const int*   __restrict__ edge_i,  // edge_index row 0
    const int*   __restrict__ edge_j,  // edge_index row 1
    const float* __restrict__ shifts,  // [E,3]
    float* __restrict__ out_ei,        // [2E]
    float* __restrict__ out_w,         // [E]
    float* __restrict__ out_vec,       // [E,3]
    float* __restrict__ out_sh,        // [E,3]
    int E, int nbox_f)
{
    __shared__ float sbox[160];  // up to 16 boxes * 9 floats

    const int t = threadIdx.x;

    // ---- stage the (tiny) per-structure cell matrices into LDS ----
    // CDNA5 path: async global->LDS DMA, tracked by ASYNCcnt.
#if __has_builtin(__builtin_amdgcn_global_load_async_to_lds_b32)
    if (t < nbox_f) {
        gl_int*  g = (gl_int*)(box + t);       // C-style cast: const-strip + addrspacecast
        lds_int* l = (lds_int*)(&sbox[t]);
        __builtin_amdgcn_global_load_async_to_lds_b32(g, l, 0, 0);
    }
#if __has_builtin(__builtin_amdgcn_s_wait_asynccnt)
    __builtin_amdgcn_s_wait_asynccnt(0);
#else
    asm volatile("s_wait_asynccnt 0x0" ::: "memory");
#endif
#else
    if (t < nbox_f) sbox[t] = box[t];
#endif
    __syncthreads();

    const int stride = (int)(gridDim.x * blockDim.x);
    for (int e = (int)(blockIdx.x * blockDim.x) + t; e < E; e += stride) {
        // prefetch next grid-stride iteration's shift triple (global_prefetch_b8)
        int en = e + stride;
        if (en < E) __builtin_prefetch(shifts + 3 * (size_t)en, 0, 0);

        const int i = edge_i[e];
        const int j = edge_j[e];
        const int b = batch[i];

        const Float3 s  = *(const Float3*)(shifts + 3 * (size_t)e);
        const Float3 pi = *(const Float3*)(pos    + 3 * (size_t)i);
        const Float3 pj = *(const Float3*)(pos    + 3 * (size_t)j);

        // cell_shifts[m] = sum_n s[n] * box[b][n][m]   (row-vector x matrix)
        const float* c = &sbox[b * 9];
        const float cs0 = fmaf(s.x, c[0], fmaf(s.y, c[3], s.z * c[6]));
        const float cs1 = fmaf(s.x, c[1], fmaf(s.y, c[4], s.z * c[7]));
        const float cs2 = fmaf(s.x, c[2], fmaf(s.y, c[5], s.z * c[8]));

        const float dx = pj.x - pi.x + cs0;
        const float dy = pj.y - pi.y + cs1;
        const float dz = pj.z - pi.z + cs2;

        const float w = sqrtf(fmaf(dx, dx, fmaf(dy, dy, dz * dz)));

        // outputs (tuple flattened to float): edge_index | weight | vec | shifts
        out_ei[e]              = (float)i;
        out_ei[(size_t)E + e]  = (float)j;
        out_w[e]               = w;

        Float3 v; v.x = -dx; v.y = -dy; v.z = -dz;
        *(Float3*)(out_vec + 3 * (size_t)e) = v;
        *(Float3*)(out_sh  + 3 * (size_t)e) = s;
    }
}

extern "C" void kernel_launch(void* const* d_in, const int* in_sizes, int n_in,
                              void* d_out, int out_size, void* d_ws, size_t ws_size,
                              hipStream_t stream) {
    // setup_inputs() order: pos, box, batch, edge_index, shifts_idx
    const float* pos    = (const float*)d_in[0];
    const float* box    = (const float*)d_in[1];
    const int*   batch  = (const int*)d_in[2];
    const int*   eidx   = (const int*)d_in[3];
    const float* shifts = (const float*)d_in[4];

    const int E      = in_sizes[3] / 2;   // edge_index is [2,E]
    const int nbox_f = in_sizes[1];       // B*9 floats (<=144)

    float* out     = (float*)d_out;
    float* out_ei  = out;                    // 2E
    float* out_w   = out + 2 * (size_t)E;    // E
    float* out_vec = out + 3 * (size_t)E;    // 3E
    float* out_sh  = out + 6 * (size_t)E;    // 3E

    const int threads = BPD_BLOCK;
    int blocks = (E + threads - 1) / threads;
    if (blocks > 4096) blocks = 4096;
    if (blocks < 1) blocks = 1;

    hipLaunchKernelGGL(bpd_edges_kernel, dim3(blocks), dim3(threads), 0, stream,
                       pos, box, batch,
                       eidx, eidx + (size_t)E, shifts,
                       out_ei, out_w, out_vec, out_sh,
                       E, nbox_f);
}